// SeparatedAttention_22144851378575
// MI455X (gfx1250) — compile-verified
//
#include <hip/hip_runtime.h>
#include <hip/hip_bf16.h>

// ---------------------------------------------------------------------------
// SeparatedAttention for MI455X (gfx1250), fp32 end-to-end.
// All matrix math uses V_WMMA_F32_16X16X4_F32 (wave32, 16x16 f32 C/D tile,
// A = 16x4 f32 -> 2 VGPRs/lane, B = 4x16 f32 -> 2 VGPRs/lane).
// A and B fragments are both single aligned ds_load_b64 ops: A panels are
// stored row-major [M][K], B panels transposed [N][K], with padded strides
// to spread fragment lanes across LDS banks.
//
// Problem constants: B=64, N=825, C=1024, H=16, d=64, VAT=57, ACT=56,
//                    NG=3, gs=256, nna=768
//
// Workspace layout (floats), total ~368.3M floats (~1.48 GB):
//   qkv_non  [B*768, 3072]
//   attn_non [B*768, 1024]
//   q_act    [B*57 , 1024]
//   kv_act   [B*825, 2048]
//   sx       [B*H, 64, 784]   (cross-attn scores, padded 57->64, 769->784)
//   attn_act [B*57 , 1024]
// ---------------------------------------------------------------------------

typedef __attribute__((ext_vector_type(2))) float v2f;
typedef __attribute__((ext_vector_type(8))) float v8f;

static __device__ __forceinline__ v8f wmma4(v2f a, v2f b, v8f c) {
  // D = A(16x4) * B(4x16) + C, fp32.
  // (neg_a, A, neg_b, B, c_mod, C, reuse_a, reuse_b)
  return __builtin_amdgcn_wmma_f32_16x16x4_f32(
      false, a, false, b, (short)0, c, false, false);
}

// ---------------------------------------------------------------------------
// Generic WMMA-f32 GEMM:  C[M,N] = A'[M,K] @ B[K,N] (+ bias)
// Row remap (handles x slicing and output concat):
//   physical_row(r) = off + r + (r / rpb) * gap
// M % 64 == 0, N % 128 == 0, K % 32 == 0 for all instances here.
// Block: 256 threads = 8 waves, 2(M) x 4(N); each wave owns a 32x32 tile
// (4 accumulators) -> 4 WMMAs per 4 fragment loads.
// Workgroup tile: 64(M) x 128(N), K staged in LDS in chunks of 32.
// ---------------------------------------------------------------------------
#define GT_M 64
#define GT_N 128
#define GT_K 32
#define LDA_S 34   // padded LDS stride for A rows (even -> b64-aligned)
#define LDB_S 34   // padded LDS stride for Bt rows

__global__ __launch_bounds__(256) void gemm_wmma_f32_kernel(
    const float* __restrict__ A, const float* __restrict__ Bm,
    float* __restrict__ Cm, const float* __restrict__ bias,
    int K, int lda, int ldb, int ldc,
    int aRpb, int aGap, int aOff,
    int cRpb, int cGap, int cOff) {
  __shared__ float Asl[GT_M * LDA_S];    // 64x32 panel, row-major [M][K]
  __shared__ float Bsl[GT_N * LDB_S];    // 32x128 panel, transposed [N][K]

  const int tid  = threadIdx.x;
  const int wv   = tid >> 5;
  const int lane = tid & 31;
  const int l16  = lane & 15;
  const int hl   = lane >> 4;            // half-wave: K pair / M block select

  const int m0 = blockIdx.x * GT_M;
  const int n0 = blockIdx.y * GT_N;
  const int wm = (wv & 1) * 32;          // wave M offset inside tile
  const int wn = (wv >> 1) * 32;         // wave N offset inside tile

  v8f acc00 = {}, acc01 = {}, acc10 = {}, acc11 = {};

  for (int k0 = 0; k0 < K; k0 += GT_K) {
    // Stage A tile (64x32): 2048 elems, 8 per thread, coalesced on K.
    #pragma unroll
    for (int i = tid; i < GT_M * GT_K; i += 256) {
      const int r = i >> 5;              // /GT_K
      const int c = i & 31;
      const int gr = m0 + r;
      const int ar = aOff + gr + (gr / aRpb) * aGap;
      Asl[r * LDA_S + c] = A[(size_t)ar * lda + (k0 + c)];
    }
    // Stage B tile (32x128) transposed: 4096 elems, 16 per thread,
    // coalesced on N in global, strided (2-way max) into LDS.
    #pragma unroll
    for (int i = tid; i < GT_K * GT_N; i += 256) {
      const int r = i >> 7;              // k within chunk
      const int c = i & 127;             // n within tile
      Bsl[c * LDB_S + r] = Bm[(size_t)(k0 + r) * ldb + (n0 + c)];
    }
    // Hint next K-tile into cache (global_prefetch_b8).
    if (k0 + GT_K < K) {
      const int gr = m0 + (tid >> 2);
      const int ar = aOff + gr + (gr / aRpb) * aGap;
      __builtin_prefetch(&A[(size_t)ar * lda + (k0 + GT_K)], 0, 1);
      __builtin_prefetch(&Bm[(size_t)(k0 + GT_K + (tid >> 3)) * ldb + n0 + (tid & 7) * 16], 0, 1);
    }
    __syncthreads();

    #pragma unroll
    for (int kk = 0; kk < GT_K; kk += 4) {
      // lanes 0-15 -> K=kk..kk+1, lanes 16-31 -> K=kk+2..kk+3
      v2f a0 = *(const v2f*)&Asl[(wm + l16)      * LDA_S + kk + 2 * hl];
      v2f a1 = *(const v2f*)&Asl[(wm + 16 + l16) * LDA_S + kk + 2 * hl];
      v2f b0 = *(const v2f*)&Bsl[(wn + l16)      * LDB_S + kk + 2 * hl];
      v2f b1 = *(const v2f*)&Bsl[(wn + 16 + l16) * LDB_S + kk + 2 * hl];
      acc00 = wmma4(a0, b0, acc00);
      acc01 = wmma4(a0, b1, acc01);
      acc10 = wmma4(a1, b0, acc10);
      acc11 = wmma4(a1, b1, acc11);
    }
    __syncthreads();
  }

  // C/D layout: VGPR r -> M=r (lanes 0-15) / M=8+r (lanes 16-31), N=l16.
  #pragma unroll
  for (int r = 0; r < 8; ++r) {
    const int gm0 = m0 + wm + r + 8 * hl;
    const int gm1 = gm0 + 16;
    const int gn0 = n0 + wn + l16;
    const int gn1 = gn0 + 16;
    const int cr0 = cOff + gm0 + (gm0 / cRpb) * cGap;
    const int cr1 = cOff + gm1 + (gm1 / cRpb) * cGap;
    float v00 = acc00[r], v01 = acc01[r], v10 = acc10[r], v11 = acc11[r];
    if (bias) {
      const float bz0 = bias[gn0], bz1 = bias[gn1];
      v00 += bz0; v01 += bz1; v10 += bz0; v11 += bz1;
    }
    Cm[(size_t)cr0 * ldc + gn0] = v00;
    Cm[(size_t)cr0 * ldc + gn1] = v01;
    Cm[(size_t)cr1 * ldc + gn0] = v10;
    Cm[(size_t)cr1 * ldc + gn1] = v11;
  }
}

// ---------------------------------------------------------------------------
// Grouped windowed self-attention over the 768 non-act tokens.
// One block handles (b, g, h, half): 128 queries vs all 256 keys of a group.
// Dynamic LDS (~292.5 KB of the 320 KB WGP pool):
//   Ks[256 x 66]  (row-major [key][d], padded)
//   Vt[64  x 258] (transposed [d][key], padded)
//   Qs[128 x 66]
//   Ss[128 x 258]
// 256 threads = 8 waves; wave w owns query rows [w*16, w*16+16).
// ---------------------------------------------------------------------------
#define SA_LK 66     // Ks/Qs row stride
#define SA_LS 258    // Ss/Vt row stride

__global__ __launch_bounds__(256) void selfattn_wmma_kernel(
    const float* __restrict__ qkv,   // [B*768, 3072]
    float* __restrict__ o) {         // [B*768, 1024]
  extern __shared__ float sm[];
  float* Ks = sm;                      // 256*66 = 16896
  float* Vt = Ks + 256 * SA_LK;        // 64*258 = 16512
  float* Qs = Vt + 64 * SA_LS;         // 128*66 =  8448
  float* Ss = Qs + 128 * SA_LK;        // 128*258= 33024

  const int blk  = blockIdx.x;         // B*NG*H*2 = 6144
  const int half = blk & 1;
  const int h    = (blk >> 1) & 15;
  const int g    = ((blk >> 1) >> 4) % 3;
  const int b    = (blk >> 1) / 48;

  const int tid  = threadIdx.x;
  const int wv   = tid >> 5;
  const int lane = tid & 31;
  const int l16  = lane & 15;
  const int hl   = lane >> 4;

  const int row0  = b * 768 + g * 256;   // group token base in [B*768]
  const int qrow0 = half * 128;          // this block's query half
  const size_t ldq = 3072;
  const size_t baseQ = (size_t)(row0 + qrow0) * ldq + 0 * 1024 + h * 64;
  const size_t baseK = (size_t)row0 * ldq + 1 * 1024 + h * 64;
  const size_t baseV = (size_t)row0 * ldq + 2 * 1024 + h * 64;

  for (int i = tid; i < 256 * 64; i += 256) {
    const int r = i >> 6, c = i & 63;     // key, d  (coalesced on d)
    Ks[r * SA_LK + c] = qkv[baseK + (size_t)r * ldq + c];
    Vt[c * SA_LS + r] = qkv[baseV + (size_t)r * ldq + c];  // transpose
  }
  for (int i = tid; i < 128 * 64; i += 256) {
    const int r = i >> 6, c = i & 63;
    Qs[r * SA_LK + c] = qkv[baseQ + (size_t)r * ldq + c];
  }
  __syncthreads();

  const float scale = 0.125f;   // 64^-0.5

  // S = Q @ K^T * scale  -> Ss[128 x 256]
  for (int nt = 0; nt < 16; ++nt) {
    v8f acc = {};
    #pragma unroll
    for (int kk = 0; kk < 64; kk += 4) {
      v2f a  = *(const v2f*)&Qs[(wv * 16 + l16) * SA_LK + kk + 2 * hl];
      v2f bf = *(const v2f*)&Ks[(nt * 16 + l16) * SA_LK + kk + 2 * hl];
      acc = wmma4(a, bf, acc);
    }
    #pragma unroll
    for (int r = 0; r < 8; ++r)
      Ss[(wv * 16 + r + 8 * hl) * SA_LS + nt * 16 + l16] = acc[r] * scale;
  }
  __syncthreads();

  // Row softmax over 256 keys; one row per thread (threads 0..127).
  if (tid < 128) {
    float* row = Ss + tid * SA_LS;
    float m = -3.4e38f;
    for (int c = 0; c < 256; ++c) m = fmaxf(m, row[c]);
    float s = 0.f;
    for (int c = 0; c < 256; ++c) { const float e = __expf(row[c] - m); row[c] = e; s += e; }
    const float inv = 1.f / s;
    for (int c = 0; c < 256; ++c) row[c] *= inv;
  }
  __syncthreads();

  // O = P @ V  -> [128 x 64] for this half; V is transposed in LDS.
  for (int nt = 0; nt < 4; ++nt) {
    v8f acc = {};
    #pragma unroll 8
    for (int kk = 0; kk < 256; kk += 4) {
      v2f a  = *(const v2f*)&Ss[(wv * 16 + l16) * SA_LS + kk + 2 * hl];
      v2f bf = *(const v2f*)&Vt[(nt * 16 + l16) * SA_LS + kk + 2 * hl];
      acc = wmma4(a, bf, acc);
    }
    #pragma unroll
    for (int r = 0; r < 8; ++r) {
      const int q = qrow0 + wv * 16 + r + 8 * hl;
      o[(size_t)(row0 + q) * 1024 + h * 64 + nt * 16 + l16] = acc[r];
    }
  }
}

// ---------------------------------------------------------------------------
// Cross-attention scores: S[b,h] = Qact(57->64 pad) @ Kact^T (769->784 pad)
// One block per (b,h); the whole 769x64 K panel is staged in dynamic LDS
// (row-major [key][d], zero-padded to 784 keys) -> fragments are b64 loads.
// 8 waves stride over the 4x49 output tiles.
// ---------------------------------------------------------------------------
#define XS_LK 66   // K/Q panel row stride

__global__ __launch_bounds__(256) void xattn_scores_kernel(
    const float* __restrict__ qact,    // [B*57, 1024]
    const float* __restrict__ kvact,   // [B*825, 2048]
    float* __restrict__ s) {           // [B*H, 64, 784]
  extern __shared__ float sm[];
  float* Ksh = sm;                     // 784*66 = 51744
  float* Qs  = Ksh + 784 * XS_LK;      //  64*66 =  4224

  const int bh = blockIdx.x;
  const int h  = bh & 15;
  const int b  = bh >> 4;
  const int tid  = threadIdx.x;
  const int wv   = tid >> 5;
  const int lane = tid & 31;
  const int l16  = lane & 15;
  const int hl   = lane >> 4;

  const size_t kbase = ((size_t)b * 825 + 56) * 2048 + h * 64;  // K: kv idx 0

  for (int i = tid; i < 64 * 64; i += 256) {
    const int r = i >> 6, c = i & 63;
    Qs[r * XS_LK + c] =
        (r < 57) ? qact[(size_t)(b * 57 + r) * 1024 + h * 64 + c] : 0.f;
  }
  for (int i = tid; i < 784 * 64; i += 256) {
    const int k = i >> 6, c = i & 63;   // key, d (coalesced on d)
    Ksh[k * XS_LK + c] =
        (k < 769) ? kvact[kbase + (size_t)k * 2048 + c] : 0.f;
  }
  __syncthreads();

  const float scale = 0.125f;
  const size_t sbase = (size_t)bh * 64 * 784;

  for (int t = wv; t < 4 * 49; t += 8) {
    const int mt = t & 3;
    const int nt = t >> 2;
    v8f acc = {};
    #pragma unroll
    for (int kk = 0; kk < 64; kk += 4) {
      v2f a  = *(const v2f*)&Qs[(mt * 16 + l16) * XS_LK + kk + 2 * hl];
      v2f bf = *(const v2f*)&Ksh[(nt * 16 + l16) * XS_LK + kk + 2 * hl];
      acc = wmma4(a, bf, acc);
    }
    #pragma unroll
    for (int r = 0; r < 8; ++r)
      s[sbase + (size_t)(mt * 16 + r + 8 * hl) * 784 + nt * 16 + l16] =
          acc[r] * scale;
  }
}

// Row softmax over the 769 valid keys; zero padded columns; zero pad rows.
__global__ __launch_bounds__(256) void xattn_softmax_kernel(
    float* __restrict__ s) {           // [B*H, 64, 784]
  const int bh = blockIdx.x;
  const size_t base = (size_t)bh * 64 * 784;
  const int wv   = threadIdx.x >> 5;
  const int lane = threadIdx.x & 31;

  for (int r = wv; r < 57; r += 8) {
    float* row = s + base + (size_t)r * 784;
    float m = -3.4e38f;
    for (int c = lane; c < 769; c += 32) m = fmaxf(m, row[c]);
    #pragma unroll
    for (int off = 16; off; off >>= 1) m = fmaxf(m, __shfl_xor(m, off, 32));
    float sum = 0.f;
    for (int c = lane; c < 769; c += 32) {
      const float e = __expf(row[c] - m);
      row[c] = e;
      sum += e;
    }
    #pragma unroll
    for (int off = 16; off; off >>= 1) sum += __shfl_xor(sum, off, 32);
    const float inv = 1.f / sum;
    for (int c = lane; c < 784; c += 32) row[c] = (c < 769) ? row[c] * inv : 0.f;
  }
  for (int r = 57 + wv; r < 64; r += 8)
    for (int c = lane; c < 784; c += 32) s[base + (size_t)r * 784 + c] = 0.f;
}

// ---------------------------------------------------------------------------
// O = P @ Vact : [64 x 784pad] @ [769 x 64] -> store rows < 57.
// V panel staged transposed in dynamic LDS: Vt[d][key], 64 x 784 (padded),
// ~196.5 KB -> fragments are single b64 loads; P rows read as float2 gathers
// from global (pad columns of P are zero, so padded keys contribute nothing).
// ---------------------------------------------------------------------------
#define XP_LV 786   // Vt row stride

__global__ __launch_bounds__(128) void xattn_pv_kernel(
    const float* __restrict__ s,       // [B*H, 64, 784]
    const float* __restrict__ kvact,   // [B*825, 2048]
    float* __restrict__ o) {           // [B*57, 1024]
  extern __shared__ float sm[];
  float* Vt = sm;                      // 64 * 786 = 50304 floats

  const int bh = blockIdx.x;
  const int h  = bh & 15;
  const int b  = bh >> 4;
  const int tid  = threadIdx.x;
  const int wv   = tid >> 5;            // 4 waves: one 16-row M tile each
  const int lane = tid & 31;
  const int l16  = lane & 15;
  const int hl   = lane >> 4;

  const size_t sbase = (size_t)bh * 64 * 784;
  const size_t vbase = ((size_t)b * 825 + 56) * 2048 + 1024 + h * 64;  // kv idx 1

  for (int i = tid; i < 784 * 64; i += 128) {
    const int k = i >> 6, c = i & 63;   // key, d (coalesced on d)
    Vt[c * XP_LV + k] =
        (k < 769) ? kvact[vbase + (size_t)k * 2048 + c] : 0.f;
  }
  __syncthreads();

  for (int nt = 0; nt < 4; ++nt) {
    v8f acc = {};
    #pragma unroll 8
    for (int kk = 0; kk < 784; kk += 4) {
      v2f a  = *(const v2f*)&s[sbase + (size_t)(wv * 16 + l16) * 784 + kk + 2 * hl];
      v2f bf = *(const v2f*)&Vt[(nt * 16 + l16) * XP_LV + kk + 2 * hl];
      acc = wmma4(a, bf, acc);
    }
    #pragma unroll
    for (int r = 0; r < 8; ++r) {
      const int q = wv * 16 + r + 8 * hl;
      if (q < 57)
        o[(size_t)(b * 57 + q) * 1024 + h * 64 + nt * 16 + l16] = acc[r];
    }
  }
}

// ---------------------------------------------------------------------------
// Host launch
// ---------------------------------------------------------------------------
extern "C" void kernel_launch(void* const* d_in, const int* in_sizes, int n_in,
                              void* d_out, int out_size, void* d_ws, size_t ws_size,
                              hipStream_t stream) {
  const float* x          = (const float*)d_in[0];  // [64, 825, 1024]
  const float* W_qkv_non  = (const float*)d_in[1];  // [1024, 3072]
  const float* W_q_act    = (const float*)d_in[2];  // [1024, 1024]
  const float* W_kv_act   = (const float*)d_in[3];  // [1024, 2048]
  const float* W_proj_non = (const float*)d_in[4];  // [1024, 1024]
  const float* b_proj_non = (const float*)d_in[5];  // [1024]
  const float* W_proj_act = (const float*)d_in[6];  // [1024, 1024]
  const float* b_proj_act = (const float*)d_in[7];  // [1024]
  float* out = (float*)d_out;                       // [64, 825, 1024]

  // Workspace partition (floats).
  float* ws = (float*)d_ws;
  float* qkv_non  = ws;                                  // 150,994,944
  float* attn_non = qkv_non  + (size_t)49152 * 3072;     //  50,331,648
  float* q_act    = attn_non + (size_t)49152 * 1024;     //   3,735,552
  float* kv_act   = q_act    + (size_t)3648  * 1024;     // 108,134,400
  float* sx       = kv_act   + (size_t)52800 * 2048;     //  51,380,224
  float* attn_act = sx       + (size_t)1024  * 64 * 784; //   3,735,552

  // 1) qkv_non = x[:, 57:] @ W_qkv_non      [49152 x 3072], K=1024
  gemm_wmma_f32_kernel<<<dim3(768, 24), 256, 0, stream>>>(
      x, W_qkv_non, qkv_non, nullptr,
      1024, 1024, 3072, 3072,
      /*aRpb*/ 768, /*aGap*/ 57, /*aOff*/ 57,
      /*cRpb*/ 1, /*cGap*/ 0, /*cOff*/ 0);

  // 2) q_act = x[:, :57] @ W_q_act          [3648 x 1024], K=1024
  gemm_wmma_f32_kernel<<<dim3(57, 8), 256, 0, stream>>>(
      x, W_q_act, q_act, nullptr,
      1024, 1024, 1024, 1024,
      /*aRpb*/ 57, /*aGap*/ 768, /*aOff*/ 0,
      /*cRpb*/ 1, /*cGap*/ 0, /*cOff*/ 0);

  // 3) kv_act = x @ W_kv_act                [52800 x 2048], K=1024
  gemm_wmma_f32_kernel<<<dim3(825, 16), 256, 0, stream>>>(
      x, W_kv_act, kv_act, nullptr,
      1024, 1024, 2048, 2048,
      /*aRpb*/ 1, /*aGap*/ 0, /*aOff*/ 0,
      /*cRpb*/ 1, /*cGap*/ 0, /*cOff*/ 0);

  // 4) grouped self-attention (B*NG*H*2 blocks, ~292.5 KB dynamic LDS)
  selfattn_wmma_kernel<<<dim3(64 * 3 * 16 * 2), 256,
                         (256 * SA_LK + 64 * SA_LS + 128 * SA_LK +
                          128 * SA_LS) * sizeof(float), stream>>>(
      qkv_non, attn_non);

  // 5-7) cross-attention
  xattn_scores_kernel<<<dim3(64 * 16), 256,
                        (784 * XS_LK + 64 * XS_LK) * sizeof(float), stream>>>(
      q_act, kv_act, sx);
  xattn_softmax_kernel<<<dim3(64 * 16), 256, 0, stream>>>(sx);
  xattn_pv_kernel<<<dim3(64 * 16), 128, 64 * XP_LV * sizeof(float), stream>>>(
      sx, kv_act, attn_act);

  // 8) out[:, :57] = attn_act @ W_proj_act + b_proj_act
  gemm_wmma_f32_kernel<<<dim3(57, 8), 256, 0, stream>>>(
      attn_act, W_proj_act, out, b_proj_act,
      1024, 1024, 1024, 1024,
      /*aRpb*/ 1, /*aGap*/ 0, /*aOff*/ 0,
      /*cRpb*/ 57, /*cGap*/ 768, /*cOff*/ 0);

  // 9) out[:, 57:] = attn_non @ W_proj_non + b_proj_non
  gemm_wmma_f32_kernel<<<dim3(768, 8), 256, 0, stream>>>(
      attn_non, W_proj_non, out, b_proj_non,
      1024, 1024, 1024, 1024,
      /*aRpb*/ 1, /*aGap*/ 0, /*aOff*/ 0,
      /*cRpb*/ 768, /*cGap*/ 57, /*cOff*/ 57);
}